// Ours_39256001085746
// MI455X (gfx1250) — compile-verified
//
#include <hip/hip_runtime.h>
#include <math.h>

// ---- static model config (mirrors reference) ----
#define BB    32
#define TT    12
#define NNODE 500
#define FINC  2
#define UU    32
#define NDm   32
#define NRES  32
#define NDIL  32
#define NSKIP 256
#define NEND  512
#define NPRED 12
#define LLAY  8
#define KS    2
#define RF    13
#define GCIN  160

typedef float v2f __attribute__((ext_vector_type(2)));
typedef float v8f __attribute__((ext_vector_type(8)));

static __device__ __forceinline__ int imin(int a, int b) { return a < b ? a : b; }

// ============================================================================
// Batched f32 GEMM on the CDNA5 matrix pipe: C[b](MxN,ldc) = A[b](MxK,lda) @
// B[b](KxN,ldb) [+ bias[m]] [+ C_old] [relu].  K must be a multiple of 4 and
// lda/ldb even (all call sites satisfy this).
// One wave computes a 32x64 strip of C (2 M-tiles x 4 N-tiles) with
// V_WMMA_F32_16X16X4_F32.  Fragment layouts per ISA 7.12.2 (wave32):
//   A 16x4 : lanes 0-15 rows M with K{0,1}; lanes 16-31 rows M with K{2,3}
//   C 16x16: lane n = l&15, VGPR r -> row r + 8*(l>>4)
// Out-of-range rows/cols use CLAMPED addresses (their results are never
// stored; C[m][n] depends only on row m / col n, so clamping is safe) which
// keeps the hot loop branch-free and EXEC all-ones for WMMA.
// flags: bit0 = accumulate from existing C, bit1 = relu on output
// ============================================================================
__global__ void __launch_bounds__(128)
gemm_f32_wmma(const float* __restrict__ A,
              const float* __restrict__ Bm,
              float* __restrict__ Cm,
              const float* __restrict__ bias,
              int M, int Nn, int K,
              int lda, int ldb, int ldc,
              long sA, long sB, long sC, int flags)
{
  const long bz = blockIdx.z;
  A  += bz * sA;
  Bm += bz * sB;
  Cm += bz * sC;

  const int lane = threadIdx.x & 31;
  const int wave = threadIdx.x >> 5;
  const int m0   = (blockIdx.x * 4 + wave) * 32;   // two 16-row tiles per wave
  const int n0   = blockIdx.y * 64;                // four 16-col tiles per wave
  if (m0 >= M) return;                             // wave-uniform branch

  const int half = lane >> 4;
  const int ml   = lane & 15;
  const int nc   = lane & 15;
  const int koff = half ? 2 : 0;

  // clamped A-row pointers for the two M-tiles
  const int r0 = imin(m0 + ml, M - 1);
  const int r1 = imin(m0 + 16 + ml, M - 1);
  const float* Ar0 = A + (long)r0 * lda + koff;
  const float* Ar1 = A + (long)r1 * lda + koff;

  // clamped B columns for the four N-tiles
  int ncol[4];
#pragma unroll
  for (int j = 0; j < 4; ++j) ncol[j] = imin(n0 + j * 16 + nc, Nn - 1);
  const float* Bp = Bm + (long)koff * ldb;

  // ---- accumulator init (cold path; guards fine here) ----
  v8f c0[4], c1[4];
#pragma unroll
  for (int j = 0; j < 4; ++j) {
#pragma unroll
    for (int r = 0; r < 8; ++r) {
      const int mA = m0 + r + 8 * half;
      const int mB = mA + 16;
      const int n  = n0 + j * 16 + nc;
      float iv0 = 0.0f, iv1 = 0.0f;
      if (n < Nn) {
        if (bias) {
          if (mA < M) iv0 = bias[mA];
          if (mB < M) iv1 = bias[mB];
        }
        if (flags & 1) {
          if (mA < M) iv0 += Cm[(long)mA * ldc + n];
          if (mB < M) iv1 += Cm[(long)mB * ldc + n];
        }
      }
      c0[j][r] = iv0;
      c1[j][r] = iv1;
    }
  }

  // ---- branch-free K loop: 2 b64 A-loads + 8 b32 B-loads -> 8 WMMAs ----
#pragma unroll 2
  for (int k0 = 0; k0 < K; k0 += 4) {
    const v2f a0 = *(const v2f*)Ar0;
    const v2f a1 = *(const v2f*)Ar1;
    Ar0 += 4;
    Ar1 += 4;
#pragma unroll
    for (int j = 0; j < 4; ++j) {
      v2f b;
      b[0] = Bp[ncol[j]];
      b[1] = Bp[ldb + ncol[j]];
      c0[j] = __builtin_amdgcn_wmma_f32_16x16x4_f32(false, a0, false, b,
                                                    (short)0, c0[j], false, false);
      c1[j] = __builtin_amdgcn_wmma_f32_16x16x4_f32(false, a1, false, b,
                                                    (short)0, c1[j], false, false);
    }
    Bp += (long)4 * ldb;
  }

  // ---- store (guards OK after the WMMA loop) ----
#pragma unroll
  for (int j = 0; j < 4; ++j) {
    const int n = n0 + j * 16 + nc;
    if (n >= Nn) continue;
#pragma unroll
    for (int r = 0; r < 8; ++r) {
      const int mA = m0 + r + 8 * half;
      const int mB = mA + 16;
      float v0 = c0[j][r];
      float v1 = c1[j][r];
      if (flags & 2) { v0 = fmaxf(v0, 0.0f); v1 = fmaxf(v1, 0.0f); }
      if (mA < M) Cm[(long)mA * ldc + n] = v0;
      if (mB < M) Cm[(long)mB * ldc + n] = v1;
    }
  }
}

static inline void gemm(hipStream_t s, const float* A, const float* B, float* C,
                        const float* bias, int M, int Nn, int K,
                        int lda, int ldb, int ldc,
                        long sA, long sB, long sC, int batch, int flags)
{
  dim3 grid((unsigned)((M + 127) / 128), (unsigned)((Nn + 63) / 64), (unsigned)batch);
  gemm_f32_wmma<<<grid, dim3(128), 0, s>>>(A, B, C, bias, M, Nn, K,
                                           lda, ldb, ldc, sA, sB, sC, flags);
}

// ============================================================================
// v = relu(factor @ map_w + map_b)           [N, ND]
// ============================================================================
__global__ void k_v_relu(const float* __restrict__ factor,
                         const float* __restrict__ mw,
                         const float* __restrict__ mb,
                         float* __restrict__ v)
{
  int i = blockIdx.x * blockDim.x + threadIdx.x;
  if (i >= NNODE * NDm) return;
  int n = i / NDm, u = i % NDm;
  float a = mb[u];
  for (int k = 0; k < UU; ++k) a += factor[n * UU + k] * mw[k * NDm + u];
  v[i] = fmaxf(a, 0.0f);
}

// ============================================================================
// Adjacency MLP: for each (i,j): h=[v[j],v[i]] -> 64 -> relu -> 32 -> relu -> 2
// writes pre-softmax logits adj[e][i][j]
// ============================================================================
__global__ void k_adj_mlp(const float* __restrict__ v,
                          const float* __restrict__ a1w, const float* __restrict__ a1b,
                          const float* __restrict__ a2w, const float* __restrict__ a2b,
                          const float* __restrict__ a3w, const float* __restrict__ a3b,
                          float* __restrict__ adj)
{
  __shared__ float w1[64 * 64], w2[64 * 32], w3[64], b1[64], b2[32], b3[2];
  for (int t = threadIdx.x; t < 64 * 64; t += blockDim.x) w1[t] = a1w[t];
  for (int t = threadIdx.x; t < 64 * 32; t += blockDim.x) w2[t] = a2w[t];
  for (int t = threadIdx.x; t < 64; t += blockDim.x) { w3[t] = a3w[t]; b1[t] = a1b[t]; }
  for (int t = threadIdx.x; t < 32; t += blockDim.x) b2[t] = a2b[t];
  if (threadIdx.x < 2) b3[threadIdx.x] = a3b[threadIdx.x];
  __syncthreads();

  long idx = (long)blockIdx.x * blockDim.x + threadIdx.x;
  if (idx >= (long)NNODE * NNODE) return;
  int i = (int)(idx / NNODE), j = (int)(idx % NNODE);
  const float* vi = v + i * NDm;
  const float* vj = v + j * NDm;

  float h1[64];
  for (int o = 0; o < 64; ++o) {
    float a = b1[o];
    for (int k = 0; k < 32; ++k) a += vj[k] * w1[k * 64 + o];          // src half
    for (int k = 0; k < 32; ++k) a += vi[k] * w1[(32 + k) * 64 + o];   // dst half
    h1[o] = fmaxf(a, 0.0f);
  }
  float h2[32];
  for (int o = 0; o < 32; ++o) {
    float a = b2[o];
    for (int k = 0; k < 64; ++k) a += h1[k] * w2[k * 32 + o];
    h2[o] = fmaxf(a, 0.0f);
  }
  float e0 = b3[0], e1 = b3[1];
  for (int k = 0; k < 32; ++k) { e0 += h2[k] * w3[2 * k]; e1 += h2[k] * w3[2 * k + 1]; }
  adj[(long)i * NNODE + j]                       = e0;
  adj[(long)NNODE * NNODE + (long)i * NNODE + j] = e1;
}

// ============================================================================
// In-place per-row softmax over j, then max with identity: supports[e][i][j]
// ============================================================================
__global__ void k_softmax_support(float* __restrict__ sup)
{
  const int row = blockIdx.x;           // e*NNODE + i
  const int i   = row % NNODE;
  float* p = sup + (long)row * NNODE;
  __shared__ float red[256];

  int j0 = threadIdx.x, j1 = threadIdx.x + 256;
  float x0 = (j0 < NNODE) ? p[j0] : -1.0e30f;
  float x1 = (j1 < NNODE) ? p[j1] : -1.0e30f;
  red[threadIdx.x] = fmaxf(x0, x1);
  __syncthreads();
  for (int s = 128; s > 0; s >>= 1) {
    if (threadIdx.x < s) red[threadIdx.x] = fmaxf(red[threadIdx.x], red[threadIdx.x + s]);
    __syncthreads();
  }
  const float mx = red[0];
  __syncthreads();
  float e0 = (j0 < NNODE) ? __expf(x0 - mx) : 0.0f;
  float e1 = (j1 < NNODE) ? __expf(x1 - mx) : 0.0f;
  red[threadIdx.x] = e0 + e1;
  __syncthreads();
  for (int s = 128; s > 0; s >>= 1) {
    if (threadIdx.x < s) red[threadIdx.x] += red[threadIdx.x + s];
    __syncthreads();
  }
  const float inv = 1.0f / red[0];
  if (j0 < NNODE) p[j0] = fmaxf(e0 * inv, (j0 == i) ? 1.0f : 0.0f);
  if (j1 < NNODE) p[j1] = fmaxf(e1 * inv, (j1 == i) ? 1.0f : 0.0f);
}

// ============================================================================
// static-factor einsums + input assembly
// ============================================================================
__global__ void k_tmp1(const float* __restrict__ inputs,
                       const float* __restrict__ factor, float* __restrict__ tmp1)
{
  int idx = blockIdx.x * blockDim.x + threadIdx.x;   // [b][t][u][f]
  if (idx >= BB * TT * UU * FINC) return;
  int f = idx % FINC, u = (idx / FINC) % UU, t = (idx / (FINC * UU)) % TT,
      b = idx / (FINC * UU * TT);
  const float* ip = inputs + ((long)(b * TT + t) * NNODE) * FINC + f;
  float a = 0.0f;
  for (int n = 0; n < NNODE; ++n) a += ip[n * FINC] * factor[n * UU + u];
  tmp1[idx] = a;
}

__global__ void k_static(const float* __restrict__ tmp1,
                         const float* __restrict__ factor, float* __restrict__ st)
{
  int idx = blockIdx.x * blockDim.x + threadIdx.x;   // [b][t][p][f]
  if (idx >= BB * TT * NNODE * FINC) return;
  int f = idx % FINC, p = (idx / FINC) % NNODE, t = (idx / (FINC * NNODE)) % TT,
      b = idx / (FINC * NNODE * TT);
  const float* tp = tmp1 + ((long)(b * TT + t) * UU) * FINC + f;
  float a = 0.0f;
  for (int u = 0; u < UU; ++u) a += tp[u * FINC] * factor[p * UU + u];
  st[idx] = a;
}

// in4 dense [B][4][RF][N]; front-padded by RF-TT zeros in time
__global__ void k_build_in4(const float* __restrict__ inputs,
                            const float* __restrict__ st, float* __restrict__ in4)
{
  int idx = blockIdx.x * blockDim.x + threadIdx.x;
  if (idx >= BB * 4 * RF * NNODE) return;
  int n  = idx % NNODE;
  int tp = (idx / NNODE) % RF;
  int f4 = (idx / (NNODE * RF)) % 4;
  int b  = idx / (NNODE * RF * 4);
  float val = 0.0f;
  if (tp >= RF - TT) {
    int  t  = tp - (RF - TT);
    int  f  = f4 & 1;
    long si = ((long)(b * TT + t) * NNODE + n) * FINC + f;
    float s = st[si];
    val = (f4 < 2) ? s : (inputs[si] - s);
  }
  in4[idx] = val;
}

// ============================================================================
// Gated dilated temporal conv: G = tanh(Wf*X+bf) * sigmoid(Wg*X+bg)
// X dense [B][32][t_in][N], G dense [B][32][t_out][N]
// ============================================================================
__global__ void k_tconv(const float* __restrict__ X,
                        const float* __restrict__ wf, const float* __restrict__ bf,
                        const float* __restrict__ wg, const float* __restrict__ bg,
                        float* __restrict__ G, int t_in, int t_out, int d)
{
  long idx   = (long)blockIdx.x * blockDim.x + threadIdx.x;
  long total = (long)BB * NDIL * t_out * NNODE;
  if (idx >= total) return;
  int n  = (int)(idx % NNODE);
  int tt = (int)((idx / NNODE) % t_out);
  int o  = (int)((idx / ((long)NNODE * t_out)) % NDIL);
  int b  = (int)(idx / ((long)NNODE * t_out * NDIL));

  float af = bf[o], ag = bg[o];
  for (int c = 0; c < NRES; ++c) {
    const float* xp = X + (((long)(b * NRES + c) * t_in) + tt) * NNODE + n;
#pragma unroll
    for (int k = 0; k < KS; ++k) {
      float xv = xp[(long)k * d * NNODE];
      af += xv * wf[(o * NRES + c) * KS + k];
      ag += xv * wg[(o * NRES + c) * KS + k];
    }
  }
  G[idx] = tanhf(af) * (1.0f / (1.0f + __expf(-ag)));
}

// ============================================================================
// Residual add (tail-aligned) + deterministic per-channel partial sums
// ============================================================================
__global__ void k_resadd_stats(float* __restrict__ Y, const float* __restrict__ Xold,
                               float* __restrict__ psum, float* __restrict__ psq,
                               int t_out, int t_in, int chunks)
{
  const int b = blockIdx.z, c = blockIdx.y, ch = blockIdx.x;
  const int e = ch * blockDim.x + threadIdx.x;      // index within t_out*NNODE
  const int lim = t_out * NNODE;
  float y = 0.0f;
  if (e < lim) {
    int  tt = e / NNODE, n = e % NNODE;
    long yi = (((long)(b * NRES + c) * t_out) + tt) * NNODE + n;
    long xi = (((long)(b * NRES + c) * t_in) + (tt + (t_in - t_out))) * NNODE + n;
    y = Y[yi] + Xold[xi];
    Y[yi] = y;
  }
  __shared__ float s1[256], s2[256];
  s1[threadIdx.x] = (e < lim) ? y : 0.0f;
  s2[threadIdx.x] = (e < lim) ? y * y : 0.0f;
  __syncthreads();
  for (int s = 128; s > 0; s >>= 1) {
    if (threadIdx.x < s) {
      s1[threadIdx.x] += s1[threadIdx.x + s];
      s2[threadIdx.x] += s2[threadIdx.x + s];
    }
    __syncthreads();
  }
  if (threadIdx.x == 0) {
    int z = (c * BB + b) * chunks + ch;
    psum[z] = s1[0];
    psq[z]  = s2[0];
  }
}

__global__ void k_stats_final(const float* __restrict__ psum,
                              const float* __restrict__ psq,
                              float* __restrict__ stats, int nz)
{
  int c = threadIdx.x;
  if (c >= NRES) return;
  float a = 0.0f, q = 0.0f;
  for (int z = 0; z < nz; ++z) { a += psum[c * nz + z]; q += psq[c * nz + z]; }
  stats[c] = a;
  stats[NRES + c] = q;
}

__global__ void k_bn_apply(float* __restrict__ Y, const float* __restrict__ stats,
                           const float* __restrict__ g, const float* __restrict__ bta,
                           int t_out)
{
  long idx   = (long)blockIdx.x * blockDim.x + threadIdx.x;
  long total = (long)BB * NRES * t_out * NNODE;
  if (idx >= total) return;
  int c = (int)((idx / ((long)t_out * NNODE)) % NRES);
  float cnt = (float)(BB * t_out * NNODE);
  float m   = stats[c] / cnt;
  float var = stats[NRES + c] / cnt - m * m;
  Y[idx] = (Y[idx] - m) * rsqrtf(var + 1e-5f) * g[c] + bta[c];
}

__global__ void k_relu(float* __restrict__ p, long n)
{
  long i = (long)blockIdx.x * blockDim.x + threadIdx.x;
  if (i < n) p[i] = fmaxf(p[i], 0.0f);
}

// ============================================================================
// Host orchestration
// ============================================================================
extern "C" void kernel_launch(void* const* d_in, const int* in_sizes, int n_in,
                              void* d_out, int out_size, void* d_ws, size_t ws_size,
                              hipStream_t stream)
{
  (void)in_sizes; (void)n_in; (void)out_size; (void)ws_size;

  const float* inputs  = (const float*)d_in[0];
  const float* factor  = (const float*)d_in[1];
  const float* map_w   = (const float*)d_in[2];
  const float* map_b   = (const float*)d_in[3];
  const float* a1w     = (const float*)d_in[4];
  const float* a1b     = (const float*)d_in[5];
  const float* a2w     = (const float*)d_in[6];
  const float* a2b     = (const float*)d_in[7];
  const float* a3w     = (const float*)d_in[8];
  const float* a3b     = (const float*)d_in[9];
  const float* enter_w = (const float*)d_in[10];
  const float* enter_b = (const float*)d_in[11];
  const float* filt_w  = (const float*)d_in[12];
  const float* filt_b  = (const float*)d_in[13];
  const float* gate_w  = (const float*)d_in[14];
  const float* gate_b  = (const float*)d_in[15];
  const float* skp_w   = (const float*)d_in[16];
  const float* skp_b   = (const float*)d_in[17];
  const float* gc_w    = (const float*)d_in[18];
  const float* gc_b    = (const float*)d_in[19];
  const float* bn_g    = (const float*)d_in[20];
  const float* bn_b    = (const float*)d_in[21];
  const float* o1w     = (const float*)d_in[22];
  const float* o1b     = (const float*)d_in[23];
  const float* o2w     = (const float*)d_in[24];
  const float* o2b     = (const float*)d_in[25];

  float* out_y = (float*)d_out;                          // [B][NPRED][N] = 192000
  float* sup   = out_y + (long)BB * NPRED * NNODE;       // [2][N][N]    = 500000

  // ---- workspace layout (floats) ----
  const long SZX = (long)BB * NRES * RF * NNODE;         // 6,656,000
  float* ws      = (float*)d_ws;
  float* f_v     = ws;                                   // 16,000
  float* f_x     = ws + 16000;                           // ping
  float* f_y     = f_x + SZX;                            // pong
  float* f_g     = f_y + SZX;                            // gated output
  float* f_p1    = f_g + SZX;                            // A^1 diffusion
  float* f_p2    = f_p1 + SZX;                           // A^2 diffusion
  float* f_skip  = f_p2 + SZX;                           // [B][256][N]
  float* f_head  = f_skip + (long)BB * NSKIP * NNODE;    // [B][512][N]
  float* f_stats = f_head + (long)BB * NEND * NNODE;     // 64
  float* f_psum  = f_stats + 64;                         // 32*32*24
  float* f_psq   = f_psum + (long)NRES * BB * 24;
  // setup-only buffers alias the head region (head is used only after layers)
  float* f_tmp1   = f_head;                              // 24,576
  float* f_static = f_head + 24576;                      // 384,000
  float* f_in4    = f_static + 384000;                   // 832,000

  // ---- adaptive supports ----
  k_v_relu<<<(NNODE * NDm + 255) / 256, 256, 0, stream>>>(factor, map_w, map_b, f_v);
  k_adj_mlp<<<(unsigned)(((long)NNODE * NNODE + 255) / 256), 256, 0, stream>>>(
      f_v, a1w, a1b, a2w, a2b, a3w, a3b, sup);
  k_softmax_support<<<2 * NNODE, 256, 0, stream>>>(sup);

  // ---- input split + enter conv ----
  k_tmp1<<<(BB * TT * UU * FINC + 255) / 256, 256, 0, stream>>>(inputs, factor, f_tmp1);
  k_static<<<(BB * TT * NNODE * FINC + 255) / 256, 256, 0, stream>>>(f_tmp1, factor, f_static);
  k_build_in4<<<(BB * 4 * RF * NNODE + 255) / 256, 256, 0, stream>>>(inputs, f_static, f_in4);
  // x = enter_w @ in4 + enter_b : batched over b, M=32, K=4, N=RF*NNODE
  gemm(stream, enter_w, f_in4, f_x, enter_b, NRES, RF * NNODE, 4,
       4, RF * NNODE, RF * NNODE,
       0, (long)4 * RF * NNODE, (long)NRES * RF * NNODE, BB, 0);

  hipMemsetAsync(f_skip, 0, (size_t)BB * NSKIP * NNODE * sizeof(float), stream);

  // ---- stacked ST layers ----
  const int dil[LLAY] = {1, 2, 1, 2, 1, 2, 1, 2};
  int t = RF;
  float* X = f_x;
  float* Y = f_y;
  for (int i = 0; i < LLAY; ++i) {
    const int d  = dil[i];
    const int to = t - d;

    // gated temporal conv -> G [B][32][to][N]
    long tot = (long)BB * NDIL * to * NNODE;
    k_tconv<<<(unsigned)((tot + 255) / 256), 256, 0, stream>>>(
        X, filt_w + (long)i * NDIL * NRES * KS, filt_b + i * NDIL,
        gate_w + (long)i * NDIL * NRES * KS, gate_b + i * NDIL,
        f_g, t, to, d);

    // skip: only the last time column of each layer survives the truncation
    // skipAcc[b] += skp_w @ G_b[:, to-1, :] + skp_b
    gemm(stream, skp_w + (long)i * NSKIP * NDIL, f_g + (long)(to - 1) * NNODE, f_skip,
         skp_b + (long)i * NSKIP,
         NSKIP, NNODE, NDIL,
         NDIL, to * NNODE, NNODE,
         0, (long)NDIL * to * NNODE, (long)NSKIP * NNODE, BB, 1);

    // graph conv: Y = gc_b + Sum_k gcW_k @ chunk_k, chunks = [G, GS0, GS0^2, GS1, GS1^2]
    const float* gcwL = gc_w + (long)i * NRES * GCIN;
    const float* gcbL = gc_b + (long)i * NRES;
    const int    Nn2  = to * NNODE;
    const long   sG   = (long)NDIL * to * NNODE;
    const int    Mbig = BB * NDIL * to;

    gemm(stream, gcwL, f_g, Y, gcbL, NRES, Nn2, NDIL, GCIN, Nn2, Nn2, 0, sG, sG, BB, 0);
    for (int e = 0; e < 2; ++e) {
      const float* S = sup + (long)e * NNODE * NNODE;
      gemm(stream, f_g, S, f_p1, nullptr, Mbig, NNODE, NNODE,
           NNODE, NNODE, NNODE, 0, 0, 0, 1, 0);                       // P1 = G x_v S
      gemm(stream, gcwL + (1 + 2 * e) * NDIL, f_p1, Y, nullptr, NRES, Nn2, NDIL,
           GCIN, Nn2, Nn2, 0, sG, sG, BB, 1);                         // Y += W @ P1
      gemm(stream, f_p1, S, f_p2, nullptr, Mbig, NNODE, NNODE,
           NNODE, NNODE, NNODE, 0, 0, 0, 1, 0);                       // P2 = P1 x_v S
      gemm(stream, gcwL + (2 + 2 * e) * NDIL, f_p2, Y, nullptr, NRES, Nn2, NDIL,
           GCIN, Nn2, Nn2, 0, sG, sG, BB, 1);                         // Y += W @ P2
    }

    // residual add + deterministic batchnorm
    const int chunks = (to * NNODE + 255) / 256;
    dim3 rg((unsigned)chunks, NRES, BB);
    k_resadd_stats<<<rg, 256, 0, stream>>>(Y, X, f_psum, f_psq, to, t, chunks);
    k_stats_final<<<1, 32, 0, stream>>>(f_psum, f_psq, f_stats, BB * chunks);
    long totY = (long)BB * NRES * to * NNODE;
    k_bn_apply<<<(unsigned)((totY + 255) / 256), 256, 0, stream>>>(
        Y, f_stats, bn_g + i * NRES, bn_b + i * NRES, to);

    float* sw = X; X = Y; Y = sw;
    t = to;
  }

  // ---- output head ----
  k_relu<<<(unsigned)(((long)BB * NSKIP * NNODE + 255) / 256), 256, 0, stream>>>(
      f_skip, (long)BB * NSKIP * NNODE);
  gemm(stream, o1w, f_skip, f_head, o1b, NEND, NNODE, NSKIP,
       NSKIP, NNODE, NNODE, 0, (long)NSKIP * NNODE, (long)NEND * NNODE, BB, 2);
  gemm(stream, o2w, f_head, out_y, o2b, NPRED, NNODE, NEND,
       NEND, NNODE, NNODE, 0, (long)NEND * NNODE, (long)NPRED * NNODE, BB, 0);
}